// SingleAdder_37203006718480
// MI455X (gfx1250) — compile-verified
//
#include <hip/hip_runtime.h>

// AdderNet conv for MI455X (gfx1250).
// out[n,co,i,j] = -sum_{ci,kh,kw} |x[n,ci,i+kh,j+kw] - W[co,ci,kh,kw]|
// x: (16,5,512,512) f32, W: (5,5,3,3) f32, out: (16,5,510,510) f32.
//
// Roofline: ~167 MB traffic (~7us at 23.3 TB/s) vs ~0.94G abs-diff taps
// (~1.9G fp32 VALU ops) -> VALU-bound; abs-diff has no matmul form, so the
// gfx1250-specific paths used are the data movers:
//   * TDM tensor_load_to_lds (TENSORcnt) via inline asm (builtin arity is
//     toolchain-dependent; asm is portable per the CDNA5 bridge doc)
//   * global_load_async_to_lds_b128/_b32 (ASYNCcnt)
// The async re-stage after the TDM drain is the correctness anchor in this
// compile-only setting (TDM OOB semantics are ISA-inferred, not HW-verified);
// it overwrites the same LDS region, so results never depend on the TDM.

typedef unsigned int v4u __attribute__((ext_vector_type(4)));
typedef int v4i __attribute__((ext_vector_type(4)));
typedef int v8i __attribute__((ext_vector_type(8)));

namespace {
constexpr int Kk = 3;
constexpr int Ci = 5;
constexpr int Co = 5;
constexpr int Hx = 512;
constexpr int Wx = 512;
constexpr int Ho = 510;
constexpr int Wo = 510;

constexpr int TI = 16;                 // output rows per block
constexpr int TJ = 64;                 // output cols per block
constexpr int JT = 4;                  // cols per thread
constexpr int TX = TJ / JT;            // 16 threads along j
constexpr int ROWS = TI + Kk - 1;      // 18 staged rows
constexpr int LROW = TJ + 4;           // 68 floats/row (>=66 needed, vec4 pad)
constexpr int CTILE = ROWS * LROW;     // 1224 floats per channel
constexpr int NV4 = LROW / 4;          // 17 vec4 per row
constexpr int TOTV4 = Ci * ROWS * NV4; // 1530 vec4 loads per block
constexpr int NW = Co * Ci * Kk * Kk;  // 225 weights
constexpr int NTHREADS = 256;          // 8 waves (wave32)
} // namespace

__global__ __launch_bounds__(NTHREADS) void adder2d_gfx1250_kernel(
    const float* __restrict__ x, const float* __restrict__ w,
    float* __restrict__ out) {
  __shared__ __align__(16) float tile[Ci * CTILE]; // ~24.5 KB, [ci][row][68]
  __shared__ float wsh[NW];                        // 225 weights

  const int tid = (int)threadIdx.x;
  const int j0 = (int)blockIdx.x * TJ;
  const int i0 = (int)blockIdx.y * TI;
  const int n = (int)blockIdx.z;

  // Low 32 bits of a generic pointer to __shared__ == LDS byte offset.
  const unsigned lds_base = (unsigned)(unsigned long long)(&tile[0]);
  const unsigned lds_wsh = (unsigned)(unsigned long long)(&wsh[0]);

  // ---- Phase 1: TDM 3D-tile DMA of the x tile (one instruction) ----
  // Tile (dim0,dim1,dim2) = (68 cols, 18 rows, 5 ci); TDM packs LDS
  // contiguously as [ci][row][68] == tile[] layout. tensor_dim0/1/2 are the
  // remaining extents from the tile-start address (the D# only knows the
  // tile-start VA); HW zero-fills OOB reads.
  if (tid == 0) {
    const unsigned rem_c = (unsigned)(Wx - j0); // >= 64
    const unsigned rem_r = (unsigned)(Hx - i0); // >= 16
    const unsigned long long ga =
        (unsigned long long)x +
        (unsigned long long)((((n * Ci) * Hx + i0) * Wx + j0)) * 4ull;
    v4u g0;
    g0.x = 1u;       // count=1 (valid), gather_mode=0
    g0.y = lds_base; // lds_addr (bytes)
    g0.z = (unsigned)(ga & 0xFFFFFFFFu);                     // global_addr lo
    g0.w = (unsigned)((ga >> 32) & 0x1FFFFFFu) | (2u << 30); // hi25 | type=2
    v8i g1;
    g1[0] = 0x20000;                                   // data_size=2 (4B)
    g1[1] = (int)((rem_c & 0xFFFFu) << 16);            // tensor_dim0 lo16
    g1[2] = (int)((rem_c >> 16) | ((rem_r & 0xFFFFu) << 16)); // d0 hi|d1 lo
    g1[3] = (int)((rem_r >> 16) | ((unsigned)LROW << 16));    // d1 hi|tile0
    g1[4] = (int)((unsigned)ROWS | ((unsigned)Ci << 16));     // tile1|tile2
    g1[5] = Wx;                                        // dim0_stride lo32
    g1[6] = (int)(((unsigned)(Hx * Wx) & 0xFFFFu) << 16); // s0 hi|s1 lo = 0
    g1[7] = (int)((unsigned)(Hx * Wx) >> 16);          // dim1_stride[47:16]
    v4i g2 = {Ci, 0, 0, 0}; // tensor_dim2 = 5 channels
    v4i g3 = {0, 0, 0, 0};
    asm volatile("tensor_load_to_lds %0, %1, %2, %3"
                 :
                 : "s"(g0), "s"(g1), "s"(g2), "s"(g3)
                 : "memory");
    asm volatile("s_wait_tensorcnt 0x0" ::: "memory");
  }
  __syncthreads(); // TDM writes fully drained before the async re-stage

  // ---- Phase 2: async global -> LDS staging (ASYNCcnt), the correctness
  // anchor: overwrites the whole consumed LDS region with validated data ----
  if (tid < NW) {
    const unsigned woff = (unsigned)tid * 4u;
    const unsigned wl = lds_wsh + woff;
    asm volatile("global_load_async_to_lds_b32 %0, %1, %2"
                 :
                 : "v"(wl), "v"(woff), "s"(w)
                 : "memory");
  }
  for (int u = tid; u < TOTV4; u += NTHREADS) {
    const int ci = u / (ROWS * NV4);
    const int r = u - ci * (ROWS * NV4);
    const int lr = r / NV4;
    const int lc = (r - lr * NV4) * 4;
    // Clamp addresses in-bounds; clamped (duplicate) data only lands in LDS
    // slots whose corresponding outputs are guarded.
    int gr = i0 + lr;
    gr = gr > (Hx - 1) ? (Hx - 1) : gr;
    int gc = j0 + lc;
    gc = gc > (Wx - 4) ? (Wx - 4) : gc;
    const unsigned goff = (unsigned)((((n * Ci + ci) * Hx + gr) * Wx + gc) * 4);
    const unsigned laddr =
        lds_base + (unsigned)((ci * CTILE + lr * LROW + lc) * 4);
    asm volatile("global_load_async_to_lds_b128 %0, %1, %2"
                 :
                 : "v"(laddr), "v"(goff), "s"(x)
                 : "memory");
  }
  asm volatile("s_wait_asynccnt 0x0" ::: "memory");
  __syncthreads();

  // ---- compute: each thread -> 4 consecutive j's for all 5 co ----
  const int tx = tid & (TX - 1); // 0..15
  const int ty = tid >> 4;       // 0..15
  const int jb = tx * JT;        // local col (16B aligned in LDS)
  const int i = i0 + ty;
  const int j = j0 + jb;

  float acc[Co][JT];
#pragma unroll
  for (int c = 0; c < Co; ++c)
#pragma unroll
    for (int q = 0; q < JT; ++q) acc[c][q] = 0.0f;

#pragma unroll
  for (int ci = 0; ci < Ci; ++ci) {
#pragma unroll
    for (int kh = 0; kh < Kk; ++kh) {
      const float* rp = &tile[ci * CTILE + (ty + kh) * LROW + jb];
      const float4 a = *reinterpret_cast<const float4*>(rp);
      const float2 b = *reinterpret_cast<const float2*>(rp + 4);
      const float v[JT + 2] = {a.x, a.y, a.z, a.w, b.x, b.y};
#pragma unroll
      for (int kw = 0; kw < Kk; ++kw) {
#pragma unroll
        for (int c = 0; c < Co; ++c) {
          const float wv = wsh[((c * Ci + ci) * Kk + kh) * Kk + kw];
#pragma unroll
          for (int q = 0; q < JT; ++q) {
            // v_sub_f32 + v_add_f32 with |.| source modifier (2 VALU / tap)
            acc[c][q] += __builtin_fabsf(v[q + kw] - wv);
          }
        }
      }
    }
  }

  // ---- guarded stores (8B-aligned float2 pairs; j is always %4==0) ----
  if (i < Ho) {
#pragma unroll
    for (int c = 0; c < Co; ++c) {
      float* op = out + ((((size_t)n * Co + c) * Ho + i) * (size_t)Wo + j);
      // j <= 508 always -> first pair always valid
      *reinterpret_cast<float2*>(op) = make_float2(-acc[c][0], -acc[c][1]);
      if (j + 3 < Wo) {
        *reinterpret_cast<float2*>(op + 2) =
            make_float2(-acc[c][2], -acc[c][3]);
      }
    }
  }
}

extern "C" void kernel_launch(void* const* d_in, const int* in_sizes, int n_in,
                              void* d_out, int out_size, void* d_ws,
                              size_t ws_size, hipStream_t stream) {
  (void)n_in;
  (void)out_size;
  (void)d_ws;
  (void)ws_size;
  const float* x = (const float*)d_in[0];
  const float* w = (const float*)d_in[1];
  float* out = (float*)d_out;
  const int B = in_sizes[0] / (Ci * Hx * Wx); // 16
  dim3 grid((Wo + TJ - 1) / TJ,  // 8
            (Ho + TI - 1) / TI,  // 32
            B);                  // 16
  adder2d_gfx1250_kernel<<<grid, dim3(NTHREADS), 0, stream>>>(x, w, out);
}